// FARNN_S_O_45801531244718
// MI455X (gfx1250) — compile-verified
//
#include <hip/hip_runtime.h>

#define B_ 16
#define L_ 128
#define S_ 64
#define C_ 8
#define V_ 1000

typedef __attribute__((ext_vector_type(2))) float v2f;
typedef __attribute__((ext_vector_type(8))) float v8f;

// workspace layout (float offsets)
#define WS_WSUM  0
#define WS_TR    (S_*S_)
#define WS_ALPHA (WS_TR + B_*L_*S_*S_)
#define WS_BETA  (WS_ALPHA + B_*L_*S_)

// ---------------------------------------------------------------------------
// Kernel 1: wsum[s,j] = sum_c wild[c,s,j] + wwm[s,j]
// ---------------------------------------------------------------------------
__global__ void k_wsum(const float* __restrict__ wild, const float* __restrict__ wwm,
                       float* __restrict__ wsum) {
  int e = blockIdx.x * blockDim.x + threadIdx.x;
  if (e < S_ * S_) {
    float acc = wwm[e];
#pragma unroll
    for (int c = 0; c < C_; ++c) acc += wild[c * S_ * S_ + e];
    wsum[e] = acc;
  }
}

// ---------------------------------------------------------------------------
// Kernel 2: Tr[b,t] = sum_c lang[v] + wsum   (one block per token, float4)
// ---------------------------------------------------------------------------
__global__ void k_build_tr(const int* __restrict__ inp, const float* __restrict__ lang,
                           const float* __restrict__ wsum, float* __restrict__ tr) {
  const int tok = blockIdx.x;              // b*L + t
  const int v = inp[tok];
  const float4* l4 = (const float4*)lang + (size_t)v * (C_ * S_ * S_ / 4);
  const float4* w4 = (const float4*)wsum;
  float4* o4 = (float4*)tr + (size_t)tok * (S_ * S_ / 4);
  for (int i = threadIdx.x; i < S_ * S_ / 4; i += blockDim.x) {
    float4 acc = w4[i];
#pragma unroll
    for (int c = 0; c < C_; ++c) {
      float4 x = l4[c * (S_ * S_ / 4) + i];
      acc.x += x.x; acc.y += x.y; acc.z += x.z; acc.w += x.w;
    }
    o4[i] = acc;
  }
}

// ---------------------------------------------------------------------------
// Kernel 3: forward/backward sequential scans.
// blocks 0..15 = forward chain for batch b, blocks 16..31 = backward chain.
// 64 threads; Tr tile staged in LDS with 65-float rows (bank-conflict free
// for both row-major and transposed dot products).
// ---------------------------------------------------------------------------
__global__ void k_scan(const int* __restrict__ lengths, const float* __restrict__ h0,
                       const float* __restrict__ hT, const float* __restrict__ tr,
                       float* __restrict__ alpha, float* __restrict__ beta) {
  __shared__ float shT[S_ * 65];
  __shared__ float shH[S_];
  const int j = threadIdx.x;
  const bool bwd = blockIdx.x >= B_;
  const int b = bwd ? (blockIdx.x - B_) : blockIdx.x;
  const int len = lengths[b];

  if (!bwd) {
    float h = h0[j];
    shH[j] = h;
    alpha[((size_t)b * L_ + 0) * S_ + j] = h;
    for (int t = 0; t < L_; ++t) {
      const float4* src = (const float4*)(tr + ((size_t)b * L_ + t) * S_ * S_);
      for (int q = j; q < S_ * S_ / 4; q += S_) {
        float4 x = src[q];
        int row = q >> 4, col = (q & 15) * 4;
        float* d = &shT[row * 65 + col];
        d[0] = x.x; d[1] = x.y; d[2] = x.z; d[3] = x.w;
      }
      __syncthreads();
      float y = 0.f;
#pragma unroll 8
      for (int s = 0; s < S_; ++s) y += shH[s] * shT[s * 65 + j];
      y = fmaxf(y, 0.f);
      __syncthreads();
      shH[j] = y;
      if (t + 1 < L_) alpha[((size_t)b * L_ + t + 1) * S_ + j] = y;
    }
  } else {
    float h = hT[j];
    shH[j] = h;
    beta[((size_t)b * L_ + (len - 1)) * S_ + j] = h;   // beta[b,len-1] = hT
    for (int i = 0; i < L_; ++i) {
      const int p = (i < len) ? (len - 1 - i) : i;     // bwd_input position
      const float4* src = (const float4*)(tr + ((size_t)b * L_ + p) * S_ * S_);
      for (int q = j; q < S_ * S_ / 4; q += S_) {
        float4 x = src[q];
        int row = q >> 4, col = (q & 15) * 4;
        float* d = &shT[row * 65 + col];
        d[0] = x.x; d[1] = x.y; d[2] = x.z; d[3] = x.w;
      }
      __syncthreads();
      float y = 0.f;
#pragma unroll 8
      for (int s = 0; s < S_; ++s) y += shT[j * 65 + s] * shH[s];   // transposed matvec
      y = fmaxf(y, 0.f);
      __syncthreads();
      shH[j] = y;
      if (i < len - 1)      beta[((size_t)b * L_ + (len - 2 - i)) * S_ + j] = y;
      else if (i >= len)    beta[((size_t)b * L_ + i) * S_ + j] = y;
    }
  }
}

// ---------------------------------------------------------------------------
// Kernel 4: score[b,t,c] = alpha^T * relu(lang[v,c]+wild[c]) * beta
//
// One block (4 waves) per token; each wave owns two c-channels and a private
// 16x72 LDS slab (no barriers needed; LDS ops are in-order within a wave).
// Per 16-row slab: b128 global loads (lang+wild), relu in registers,
// ds_store_b128, then 4 k-tiles x 4 n-tiles of chained f32 WMMAs whose B
// operand comes from LDS. Row pitch 72 makes each ds_load_b32 conflict-free
// (2*72 = 144 ≡ 16 mod 64, so lane halves hit disjoint 16-bank ranges).
//
// WMMA formulation: A (16x4) = alpha replicated across all 16 M rows (fill
// depends only on K, per the documented 32-bit A layout); B (4x16) = relu(T)
// tile; D accumulates the s-contraction across 16 chained k-tiles; all D rows
// equal u[j] = sum_s relu(T[s,j])*alpha[s]; dot with beta + shfl_xor reduce.
// ---------------------------------------------------------------------------
#define PITCH 72
__global__ void k_score(const int* __restrict__ inp, const float* __restrict__ lang,
                        const float* __restrict__ wild, const float* __restrict__ alpha,
                        const float* __restrict__ beta, float* __restrict__ out) {
  __shared__ float shB[4 * 16 * PITCH];    // 4 waves x 16-row slab (18 KB)
  const int tok = blockIdx.x;
  const int v = inp[tok];
  const int lane = threadIdx.x & 31;
  const int wave = threadIdx.x >> 5;
  const int hi = lane >> 4;      // 0: lanes 0-15, 1: lanes 16-31
  const int ln = lane & 15;
  float* myB = &shB[wave * 16 * PITCH];

  const float* aP = alpha + (size_t)tok * S_;
  const float* bP = beta + (size_t)tok * S_;

  // A operand per k-tile kt: lane needs alpha[4*kt + 2*hi + {0,1}]
  v2f aReg[16];
#pragma unroll
  for (int kt = 0; kt < 16; ++kt) {
    const float2 t2 = *(const float2*)(aP + 4 * kt + 2 * hi);
    aReg[kt].x = t2.x; aReg[kt].y = t2.y;
  }
  float bReg[4];
#pragma unroll
  for (int n = 0; n < 4; ++n) bReg[n] = bP[n * 16 + ln];

  const size_t langBase = (size_t)v * (C_ * S_ * S_);

  for (int cc = 0; cc < 2; ++cc) {
    const int c = wave * 2 + cc;
    const float* lC = lang + langBase + (size_t)c * S_ * S_;
    const float* wC = wild + (size_t)c * S_ * S_;

    v8f acc[4];
#pragma unroll
    for (int n = 0; n < 4; ++n) acc[n] = (v8f){0.f,0.f,0.f,0.f,0.f,0.f,0.f,0.f};

    for (int chunk = 0; chunk < 4; ++chunk) {        // 16 rows (s) per slab
      const float* lRow = lC + chunk * 16 * S_;
      const float* wRow = wC + chunk * 16 * S_;
      // stage relu(lang+wild) for 16x64 slab: 8 x b128 per lane
#pragma unroll
      for (int it = 0; it < 8; ++it) {
        const int f = (lane + it * 32) * 4;          // linear float idx 0..1023
        const int row = f >> 6, col = f & 63;
        float4 a4 = *(const float4*)(lRow + f);
        float4 w4 = *(const float4*)(wRow + f);
        float4 r;
        r.x = fmaxf(a4.x + w4.x, 0.f);
        r.y = fmaxf(a4.y + w4.y, 0.f);
        r.z = fmaxf(a4.z + w4.z, 0.f);
        r.w = fmaxf(a4.w + w4.w, 0.f);
        *(float4*)(&myB[row * PITCH + col]) = r;
      }
      // consume slab: 4 global k-tiles (chunk-local rows 0..15)
#pragma unroll
      for (int kt2 = 0; kt2 < 4; ++kt2) {
        const int r0 = 4 * kt2 + 2 * hi;             // B layout: VGPRv -> K=v (+2 hi)
#pragma unroll
        for (int n = 0; n < 4; ++n) {
          v2f bb;
          bb.x = myB[r0 * PITCH + n * 16 + ln];
          bb.y = myB[(r0 + 1) * PITCH + n * 16 + ln];
          acc[n] = __builtin_amdgcn_wmma_f32_16x16x4_f32(false, aReg[chunk * 4 + kt2],
                                                         false, bb, (short)0, acc[n],
                                                         false, false);
        }
      }
    }

    // D rows all equal u[16n .. 16n+15]; lanes 16-31 duplicate lanes 0-15.
    float sc = 0.f;
#pragma unroll
    for (int n = 0; n < 4; ++n) {
      float val = acc[n][0] * bReg[n];
#pragma unroll
      for (int m = 16; m >= 1; m >>= 1) val += __shfl_xor(val, m, 32);
      sc += val;
    }
    sc *= 0.5f;   // upper half-wave duplicated the partial sums
    if (lane == 0) out[(size_t)tok * C_ + c] = sc;
  }
}

// ---------------------------------------------------------------------------
extern "C" void kernel_launch(void* const* d_in, const int* in_sizes, int n_in,
                              void* d_out, int out_size, void* d_ws, size_t ws_size,
                              hipStream_t stream) {
  const int*   inp     = (const int*)d_in[0];
  const int*   lengths = (const int*)d_in[1];
  const float* lang    = (const float*)d_in[2];
  const float* wild    = (const float*)d_in[3];
  const float* wwm     = (const float*)d_in[4];
  const float* h0      = (const float*)d_in[5];
  const float* hT      = (const float*)d_in[6];
  float* ws    = (float*)d_ws;
  float* wsum  = ws + WS_WSUM;
  float* tr    = ws + WS_TR;
  float* alpha = ws + WS_ALPHA;
  float* beta  = ws + WS_BETA;
  float* out   = (float*)d_out;

  k_wsum    <<<(S_ * S_ + 255) / 256, 256, 0, stream>>>(wild, wwm, wsum);
  k_build_tr<<<B_ * L_,               256, 0, stream>>>(inp, lang, wsum, tr);
  k_scan    <<<2 * B_,                S_,  0, stream>>>(lengths, h0, hT, tr, alpha, beta);
  k_score   <<<B_ * L_,               128, 0, stream>>>(inp, lang, wild, alpha, beta, out);
}